// AggregateVideo_25598005084626
// MI455X (gfx1250) — compile-verified
//
#include <hip/hip_runtime.h>
#include <hip/hip_bf16.h>
#include <stdint.h>

#define B_DIM 64
#define L_DIM 4096
#define D_DIM 256
#define TGT   256

typedef __attribute__((ext_vector_type(2))) float v2f;
typedef __attribute__((ext_vector_type(8))) float v8f;

// idxs[k] = min(round_half_even(k*n/256), n-1), done exactly in integers:
// k*n <= 2^20, division by 256 exact -> rem==128 is the tie case.
__device__ __forceinline__ int seg_idx(int k, int n) {
  int r = k * n;
  int q = r >> 8;
  int rem = r & 255;
  q += (rem > 128) ? 1 : ((rem == 128) ? (q & 1) : 0);
  int nm1 = n - 1;
  return q < nm1 ? q : nm1;
}

// ---------------- Kernel 1: mask compaction (prefix scan in LDS) -------------
__global__ __launch_bounds__(256) void compact_k(const uint8_t* __restrict__ mask,
                                                 int* __restrict__ n_out,
                                                 int* __restrict__ idx_out) {
  const int b = blockIdx.x;
  const int t = threadIdx.x;
  const uint8_t* mb = mask + (size_t)b * L_DIM;

  __shared__ int sc[256];

  // 16 consecutive mask bytes per thread, loaded as 4 dwords (16B aligned)
  const uint32_t* mw = (const uint32_t*)mb;
  uint32_t u[4];
#pragma unroll
  for (int i = 0; i < 4; ++i) u[i] = mw[t * 4 + i];

  uint8_t loc[16];
  int c = 0;
#pragma unroll
  for (int i = 0; i < 16; ++i) {
    loc[i] = (uint8_t)((u[i >> 2] >> ((i & 3) * 8)) & 0xFF);
    c += loc[i] ? 1 : 0;
  }
  sc[t] = c;
  __syncthreads();

  // Hillis-Steele inclusive scan over 256 per-thread counts
#pragma unroll
  for (int off = 1; off < 256; off <<= 1) {
    int add = (t >= off) ? sc[t - off] : 0;
    __syncthreads();
    sc[t] += add;
    __syncthreads();
  }

  int pos = sc[t] - c;                 // exclusive prefix
  int* ib = idx_out + (size_t)b * L_DIM;
  const int jb = t * 16;
#pragma unroll
  for (int i = 0; i < 16; ++i) {
    if (loc[i]) ib[pos++] = jb + i;
  }
  if (t == 255) n_out[b] = sc[255];
}

// ------------- Kernel 2: segmented mean via V_WMMA_F32_16X16X4_F32 -----------
// Block = 256 threads (8 waves). Block owns output rows [m0, m0+16) of batch b.
// Wave w covers output columns [32w, 32w+32) as two 16-wide N tiles.
// Per K-step: A = per-row segment weights for 4 compacted rows (16x4),
//             B = gathered feat rows (4x16 per tile), C += A*B.
__global__ __launch_bounds__(256) void agg_k(const float* __restrict__ feats,
                                             const int* __restrict__ n_arr,
                                             const int* __restrict__ idx_arr,
                                             float* __restrict__ out) {
  const int b   = blockIdx.y;
  const int m0  = blockIdx.x * 16;
  const int lane = threadIdx.x & 31;
  const int wid  = threadIdx.x >> 5;

  const int n = n_arr[b];
  const int* __restrict__ idxb = idx_arr + (size_t)b * L_DIM;
  const float* __restrict__ fb = feats + (size_t)b * L_DIM * D_DIM;

  // Per-lane output row (A-matrix M index): lanes 0-15 and 16-31 both hold M=0..15
  const int m = lane & 15;
  const int s = seg_idx(m0 + m, n);
  const int e = seg_idx(m0 + m + 1, n);
  const bool single = (e <= s);
  const int  eEff = single ? (s + 1) : e;          // single-row segment -> [s, s+1)
  const float inv = single ? 1.0f : 1.0f / (float)(e - s);

  // Compacted-row range touched by this 16-row group (uniform across block)
  const int kStart = seg_idx(m0, n);
  const int e15 = seg_idx(m0 + 15, n);
  const int e16 = seg_idx(m0 + 16, n);
  const int kEnd = (e16 > e15 + 1) ? e16 : (e15 + 1);   // <= n

  // A/B half-wave K offset: VGPR0 holds K=0 (lanes 0-15) / K=2 (lanes 16-31)
  const int kk   = (lane >> 4) << 1;
  const int col0 = wid * 32 + (lane & 15);
  const int col1 = col0 + 16;

  v8f c0 = {};
  v8f c1 = {};

  for (int kbase = kStart; kbase < kEnd; kbase += 4) {
    __builtin_prefetch(idxb + kbase + 64, 0, 1);   // global_prefetch_b8

    const int kc0 = kbase + kk;
    const int kc1 = kc0 + 1;
    // Clamp gather index for out-of-range K (weight is 0 there anyway)
    const int i0 = (kc0 < kEnd) ? kc0 : (kEnd - 1);
    const int i1 = (kc1 < kEnd) ? kc1 : (kEnd - 1);
    const int r0 = idxb[i0];
    const int r1 = idxb[i1];

    const float a0 = (kc0 >= s && kc0 < eEff) ? inv : 0.0f;
    const float a1 = (kc1 >= s && kc1 < eEff) ? inv : 0.0f;

    const float* row0 = fb + (size_t)r0 * D_DIM;
    const float* row1 = fb + (size_t)r1 * D_DIM;

    v2f A  = {a0, a1};
    v2f B0 = {row0[col0], row1[col0]};
    v2f B1 = {row0[col1], row1[col1]};

    c0 = __builtin_amdgcn_wmma_f32_16x16x4_f32(false, A, false, B0, (short)0, c0,
                                               false, false);
    c1 = __builtin_amdgcn_wmma_f32_16x16x4_f32(false, A, false, B1, (short)0, c1,
                                               false, false);
  }

  // C/D layout: VGPR v -> M = v (lanes 0-15) or v+8 (lanes 16-31), N = lane&15
  const int rowHalf = (lane >> 4) * 8;
  float* __restrict__ ob = out + ((size_t)b * TGT + m0) * D_DIM;
#pragma unroll
  for (int v = 0; v < 8; ++v) {
    const int ro = v + rowHalf;
    ob[(size_t)ro * D_DIM + col0] = c0[v];
    ob[(size_t)ro * D_DIM + col1] = c1[v];
  }
}

// ---------------------------------- Launch -----------------------------------
extern "C" void kernel_launch(void* const* d_in, const int* in_sizes, int n_in,
                              void* d_out, int out_size, void* d_ws, size_t ws_size,
                              hipStream_t stream) {
  const float*   feats = (const float*)d_in[0];        // (64, 4096, 256) f32
  const uint8_t* mask  = (const uint8_t*)d_in[1];      // (64, 4096) bool (1 byte)
  float* out = (float*)d_out;                          // (64, 256, 256) f32

  // Workspace: int n[64] then int idx[64][4096]  (1 MB + 256 B)
  int* n_ws   = (int*)d_ws;
  int* idx_ws = n_ws + B_DIM;

  compact_k<<<B_DIM, 256, 0, stream>>>(mask, n_ws, idx_ws);
  agg_k<<<dim3(TGT / 16, B_DIM), 256, 0, stream>>>(feats, n_ws, idx_ws, out);
}